// ElasticDeformation_4312147165442
// MI455X (gfx1250) — compile-verified
//
#include <hip/hip_runtime.h>
#include <stdint.h>

typedef unsigned int u32;
typedef unsigned int u32x4 __attribute__((ext_vector_type(4)));
typedef int i32x4 __attribute__((ext_vector_type(4)));
typedef int i32x8 __attribute__((ext_vector_type(8)));

#define Dm 160
#define Hm 160
#define Wm 160
#define HWm (Hm * Wm)
#define DHWm (Dm * HWm)
#define Bm 4
#define Cm 2
#define NFLOW 1536 /* 3*8*8*8 floats per batch */

__device__ __forceinline__ float reflect159(float f) {
  // mirror about 0 and 159. Valid for |f| <= 318; actual coords are
  // i + disp with |disp| <~ 17 voxels (sigma=0.05 * 79.5), so one fold
  // after abs() is exact -- no fmod needed.
  f = fabsf(f);
  return (f > 159.0f) ? (318.0f - f) : f;
}

__global__ __launch_bounds__(256)
void elastic_warp_kernel(const float* __restrict__ x,
                         const float* __restrict__ flow,
                         float* __restrict__ out) {
  __shared__ float zsl[3 * 64];  // z-interpolated coarse slice (read by math)
  __shared__ __attribute__((aligned(16))) float stdm[NFLOW]; // TDM landing pad

  // Block decomposition: 4 batches * 160 z-slices * (20 h-tiles * 5 w-tiles)
  const int t  = (int)blockIdx.x;
  const int b  = t / (Dm * 100);
  const int r  = t % (Dm * 100);
  const int dz = r / 100;         // block-uniform z slice
  const int rt = r % 100;
  const int hy = (rt / 5) * 8 + ((int)threadIdx.x >> 5);
  const int wx = (rt % 5) * 32 + ((int)threadIdx.x & 31);

  // ---- CDNA5 Tensor Data Mover: DMA this batch's coarse flow block into LDS.
  // 1D tile of 1536 fp32 elements (6144 B), single descriptor, no multicast.
  if (threadIdx.x < 32) {
    uint64_t ga = (uint64_t)(uintptr_t)(flow + (size_t)b * NFLOW);
    u32 lds_off = (u32)(uintptr_t)(&stdm[0]); // low 32 bits of generic ptr = LDS byte offset
    u32x4 g0;
    g0.x = 1u;                                              // count=1 (valid D#)
    g0.y = lds_off;                                         // lds_addr
    g0.z = (u32)ga;                                         // global_addr[31:0]
    g0.w = (u32)((ga >> 32) & 0x01FFFFFFu) | (2u << 30);    // global_addr[56:32] | type=2
    i32x8 g1;
    g1[0] = (int)(2u << 16);                 // data_size = 4 bytes; wg_mask=0
    g1[1] = (int)((u32)NFLOW << 16);         // tensor_dim0 = 1536 (low bits)
    g1[2] = (int)(1u << 16);                 // tensor_dim1 = 1
    g1[3] = (int)((u32)NFLOW << 16);         // tile_dim0 = 1536
    g1[4] = 1;                               // tile_dim1 = 1
    g1[5] = NFLOW;                           // tensor_dim0_stride
    g1[6] = 0;
    g1[7] = 0;
    i32x4 g2; g2.x = 1; g2.y = 1; g2.z = 0; g2.w = 0;          // tensor_dim2/3 = 1
    i32x4 g3; g3.x = 0; g3.y = (int)(1u << 16); g3.z = 0; g3.w = 0; // tensor_dim4 = 1
    i32x8 g4;                                                  // extra group in 6-arg form
    g4[0] = 0; g4[1] = 0; g4[2] = 0; g4[3] = 0;
    g4[4] = 0; g4[5] = 0; g4[6] = 0; g4[7] = 0;
    __builtin_amdgcn_tensor_load_to_lds(g0, g1, g2, g3, g4, 0);
    __builtin_amdgcn_s_wait_tensorcnt(0);
  }

  // ---- Coarse z coordinate is block-uniform: fold the z-lerp into a
  // per-block precompute of the 2D coarse slice (3 comps * 8 * 8).
  const float S7 = 7.0f / 159.0f;
  const float fzc = dz * S7;
  const int   z0c = min((int)fzc, 6);
  const float tzc = fzc - z0c;

  if (threadIdx.x < 192) {
    const int comp = (int)threadIdx.x >> 6;  // 0..2
    const int yx   = (int)threadIdx.x & 63;  // y*8 + x
    const float* fp = flow + (size_t)b * NFLOW + comp * 512 + z0c * 64 + yx;
    float f0 = fp[0], f1 = fp[64];
    zsl[threadIdx.x] = fmaf(tzc, f1 - f0, f0);
  }
  __syncthreads();

  // ---- Per-thread 2D (y,x) coarse bilinear on the z-interpolated slice.
  float fyc = hy * S7, fxc = wx * S7;
  int y0c = min((int)fyc, 6), x0c = min((int)fxc, 6);
  float tyc = fyc - y0c, txc = fxc - x0c;
  const int cb = y0c * 8 + x0c;

  float up[3];
#pragma unroll
  for (int comp = 0; comp < 3; ++comp) {
    const float* f = zsl + comp * 64 + cb;
    float a00 = f[0], a01 = f[1], a10 = f[8], a11 = f[9];
    float c0 = fmaf(txc, a01 - a00, a00);
    float c1 = fmaf(txc, a11 - a10, a10);
    up[comp] = fmaf(tyc, c1 - c0, c0);
  }

  // ---- Sample coordinates (align_corners=True unnormalize folds to i + up*79.5),
  // reflection padding, then clamp indices so all 8 taps are in-bounds.
  float fx = reflect159(fmaf(up[0], 79.5f, (float)wx));
  float fy = reflect159(fmaf(up[1], 79.5f, (float)hy));
  float fz = reflect159(fmaf(up[2], 79.5f, (float)dz));
  int x0 = min((int)fx, Wm - 2);
  int y0 = min((int)fy, Hm - 2);
  int z0 = min((int)fz, Dm - 2);
  float tx = fx - x0, ty = fy - y0, tz = fz - z0;

  // 32-bit offsets (max index < 2^23); one 64-bit pointer add per base.
  const int sp   = z0 * HWm + y0 * Wm + x0;
  const int oidx = dz * HWm + hy * Wm + wx;
  const float* xb = x   + (size_t)(b * Cm) * DHWm;
  float*       ob = out + (size_t)(b * Cm) * DHWm;

#pragma unroll
  for (int c = 0; c < Cm; ++c) {
    const float* p = xb + c * DHWm + sp;
    // Each x-pair is contiguous; backend merges these into global_load_b64.
    float v000 = p[0],         v001 = p[1];
    float v010 = p[Wm],        v011 = p[Wm + 1];
    float v100 = p[HWm],       v101 = p[HWm + 1];
    float v110 = p[HWm + Wm], v111 = p[HWm + Wm + 1];
    float c00 = fmaf(tx, v001 - v000, v000);
    float c01 = fmaf(tx, v011 - v010, v010);
    float c10 = fmaf(tx, v101 - v100, v100);
    float c11 = fmaf(tx, v111 - v110, v110);
    float c0  = fmaf(ty, c01 - c00, c00);
    float c1  = fmaf(ty, c11 - c10, c10);
    float res = fmaf(tz, c1 - c0, c0);
    // Non-temporal store: output is write-once 131 MB; keep the 192 MB L2
    // for the gather-reuse of the input volume instead.
    __builtin_nontemporal_store(res, ob + c * DHWm + oidx);
  }
}

extern "C" void kernel_launch(void* const* d_in, const int* in_sizes, int n_in,
                              void* d_out, int out_size, void* d_ws, size_t ws_size,
                              hipStream_t stream) {
  const float* x    = (const float*)d_in[0]; // [4,2,160,160,160] fp32
  const float* flow = (const float*)d_in[1]; // [4,3,8,8,8] fp32
  float* out        = (float*)d_out;         // [4,2,160,160,160] fp32

  // 4 batches * 160 slices * (20 h-tiles * 5 w-tiles) = 64000 blocks of 256
  dim3 grid(Bm * Dm * 100), block(256);
  hipLaunchKernelGGL(elastic_warp_kernel, grid, block, 0, stream, x, flow, out);
}